// OptimizedBilinearInterp2d_77360950935553
// MI455X (gfx1250) — compile-verified
//
#include <hip/hip_runtime.h>
#include <stdint.h>

// Bilinear 2x upsample, NHWC f32: (8,256,256,64) -> (8,512,512,64)
// Memory-bound (~671 MB min traffic -> ~29us at 23.3 TB/s). Strategy:
//   - async global->LDS staging of the two source rows (CDNA5 ASYNCcnt path)
//   - ds_load_b128 reads from LDS, VALU blend
//   - non-temporal b128 stores so the 134MB input stays resident in 192MB L2

typedef float v4f __attribute__((ext_vector_type(4)));
typedef int   v4i __attribute__((ext_vector_type(4)));

#define B_      8
#define IH_     256
#define IW_     256
#define C_      64
#define OH_     512
#define OW_     512
#define C4_     16            // float4s per pixel
#define IROW4   (IW_ * C4_)   // 4096 float4s per input row
#define OROW4   (OW_ * C4_)   // 8192 float4s per output row
#define NTHREADS 256

// The async-to-LDS builtins take int4 pointers in explicit address spaces:
//   arg0: v4i addrspace(1)* (global), arg1: v4i addrspace(3)* (LDS)
typedef __attribute__((address_space(1))) v4i gv4i;
typedef __attribute__((address_space(3))) v4i lv4i;

__device__ __forceinline__ void async_copy_b128(const void* g, void* lds) {
#if __has_builtin(__builtin_amdgcn_global_load_async_to_lds_b128)
    // per-lane 16B copy, global -> LDS, tracked by ASYNCcnt
    __builtin_amdgcn_global_load_async_to_lds_b128(
        (gv4i*)(uintptr_t)g, (lv4i*)(uintptr_t)lds, 0, 0);
#else
#warning "__builtin_amdgcn_global_load_async_to_lds_b128 unavailable; synchronous fallback in use"
    *(v4f*)lds = *(const v4f*)g;
#endif
}

__device__ __forceinline__ void wait_async_zero() {
#if __has_builtin(__builtin_amdgcn_s_wait_asynccnt)
    __builtin_amdgcn_s_wait_asynccnt(0);
#else
    asm volatile("s_wait_asynccnt 0" ::: "memory");
#endif
}

__global__ __launch_bounds__(NTHREADS, 2)
void bilinear2x_async_kernel(const float* __restrict__ x, float* __restrict__ out) {
    extern __shared__ v4f tile[];  // [2][IROW4] = 128 KB

    const int oh = blockIdx.x;     // 0..511
    const int b  = blockIdx.y;     // 0..7
    const int t  = threadIdx.x;    // 0..255

    // ---- vertical coords (wave-uniform -> SALU) ----
    // src = max((oh+0.5)*0.5 - 0.5, 0) = max(0.5*oh - 0.25, 0)
    const float srcy = fmaxf(0.5f * (float)oh - 0.25f, 0.0f);
    int y0 = (int)srcy;                       // floor, srcy >= 0, <= 255.25
    if (y0 > IH_ - 1) y0 = IH_ - 1;
    const int   y1  = (y0 + 1 < IH_) ? y0 + 1 : IH_ - 1;
    const float dy  = srcy - (float)y0;
    const float ody = 1.0f - dy;

    // ---- stage rows y0 and y1 into LDS via async copies ----
    const v4f* src0 = (const v4f*)x + (size_t)(b * IH_ + y0) * IROW4;
    const v4f* src1 = (const v4f*)x + (size_t)(b * IH_ + y1) * IROW4;
#pragma unroll
    for (int i = 0; i < IROW4 / NTHREADS; ++i) {   // 16 iterations
        const int idx = t + i * NTHREADS;          // coalesced across lanes
        async_copy_b128(src0 + idx, &tile[idx]);
        async_copy_b128(src1 + idx, &tile[IROW4 + idx]);
    }
    wait_async_zero();     // this wave's async copies landed in LDS
    __syncthreads();       // all waves' copies visible

    // ---- compute one full output row (512 px * 16 float4) ----
    v4f* dst = (v4f*)out + (size_t)(b * OH_ + oh) * OROW4;

#pragma unroll 4
    for (int j = 0; j < OROW4 / NTHREADS; ++j) {   // 32 iterations
        const int fidx = t + j * NTHREADS;         // 32 consecutive float4s/wave
        const int ow = fidx >> 4;
        const int cc = fidx & (C4_ - 1);

        const float srcx = fmaxf(0.5f * (float)ow - 0.25f, 0.0f);
        const int   x0   = (int)srcx;              // <= 255 by construction
        const int   x1   = (x0 + 1 < IW_) ? x0 + 1 : IW_ - 1;
        const float dx   = srcx - (float)x0;
        const float odx  = 1.0f - dx;

        const v4f lt = tile[x0 * C4_ + cc];            // ds_load_b128
        const v4f rt = tile[x1 * C4_ + cc];
        const v4f lb = tile[IROW4 + x0 * C4_ + cc];
        const v4f rb = tile[IROW4 + x1 * C4_ + cc];

        const v4f res = (lt * odx + rt * dx) * ody + (lb * odx + rb * dx) * dy;

        // NT store: don't let the 537MB output stream evict the L2-resident input
        __builtin_nontemporal_store(res, dst + fidx);
    }
}

extern "C" void kernel_launch(void* const* d_in, const int* in_sizes, int n_in,
                              void* d_out, int out_size, void* d_ws, size_t ws_size,
                              hipStream_t stream) {
    const float* x   = (const float*)d_in[0];
    float*       out = (float*)d_out;

    dim3 grid(OH_, B_);                     // one workgroup per (output row, batch)
    const size_t shmem = 2u * IROW4 * sizeof(v4f);   // 131072 bytes
    bilinear2x_async_kernel<<<grid, NTHREADS, shmem, stream>>>(x, out);
}